// BackDeformField_43087111914132
// MI455X (gfx1250) — compile-verified
//
#include <hip/hip_runtime.h>
#include <hip/hip_bf16.h>
#include <cstdint>

// ---- problem constants (match reference) ----
#define BD 2
#define ND 16384
#define VD 10475
#define JD 55
#define KD 6

#define CHUNK 2048          // verts resident in LDS per pass (32 KB as float4)
#define WAVES 8             // 256-thread block = 8 wave32
#define PTS_PER_BLOCK 128   // 16 points per wave

typedef __attribute__((ext_vector_type(2))) float v2f;
typedef __attribute__((ext_vector_type(8))) float v8f;

__device__ __forceinline__ void wave_lds_fence() {
  // wave32 lockstep + in-order DS pipe: waiting DScnt==0 makes prior
  // cross-lane LDS stores visible to subsequent loads from this wave.
  asm volatile("s_wait_dscnt 0" ::: "memory");
}

// branchless sorted insert of (d, gi) into ascending top-6 lists
__device__ __forceinline__ void insert6(float (&bd)[KD], int (&bi)[KD],
                                        float d, int gi) {
#pragma unroll
  for (int s = 0; s < KD; ++s) {
    const bool c = d < bd[s];
    const float td = bd[s]; const int ti = bi[s];
    bd[s] = c ? d : bd[s];
    bi[s] = c ? gi : bi[s];
    d  = c ? td : d;
    gi = c ? ti : gi;
  }
}

__global__ __launch_bounds__(256, 4)
void backdeform_knn_wmma_kernel(const float* __restrict__ x,
                                const float* __restrict__ cam,
                                const float* __restrict__ lbs,
                                const float* __restrict__ vt,
                                const float* __restrict__ pv,
                                float* __restrict__ out)
{
  __shared__ float4 vbuf[CHUNK];              // verts: x,y,z,|v|^2     (32 KB)
  __shared__ float  scr[WAVES][2][16 * 17];   // per-wave transpose scratch, 2 tiles
  __shared__ float  pbuf[WAVES][16][4];       // posed points: x,y,z,|p|^2

  const int tid  = threadIdx.x;
  const int w    = tid >> 5;
  const int lane = tid & 31;
  const int half = lane >> 4;   // 0: K rows 0..1 / cols 0..7 ; 1: K rows 2..3 / cols 8..15
  const int m    = lane & 15;   // point-in-wave / vert-in-tile index
  const int b    = blockIdx.y;
  const int ptbase = blockIdx.x * PTS_PER_BLOCK + w * 16;

  // ---- inverse ortho projection of this wave's 16 points ----
  const float invs = 1.0f / cam[b * 3 + 0];
  const float tx   = cam[b * 3 + 1];
  const float ty   = cam[b * 3 + 2];

  float px = 0.f, py = 0.f, pz = 0.f, p2 = 0.f;
  {
    const int pt = ptbase + m;
    const float* xp = x + ((size_t)b * ND + pt) * 3;
    px = xp[0] * invs - tx;
    py = xp[1] * invs - ty;
    pz = xp[2] * invs;
    p2 = px * px + py * py + pz * pz;
    if (half == 0) {
      pbuf[w][m][0] = px; pbuf[w][m][1] = py;
      pbuf[w][m][2] = pz; pbuf[w][m][3] = p2;
    }
  }
  __syncthreads();

  // ---- A matrix (16x4 f32, 2 VGPRs): row m = [-2px, -2py, -2pz, 1] ----
  v2f A;
  A.x = (half == 0) ? (-2.0f * pbuf[w][m][0]) : (-2.0f * pbuf[w][m][2]);
  A.y = (half == 0) ? (-2.0f * pbuf[w][m][1]) : 1.0f;

  // ---- running top-6 (sorted ascending) per (point m, column-half) ----
  float bestd[KD];
  int   besti[KD];
#pragma unroll
  for (int s = 0; s < KD; ++s) { bestd[s] = 3.0e38f; besti[s] = 0; }

  for (int cb = 0; cb < VD; cb += CHUNK) {
    int vleft = VD - cb; if (vleft > CHUNK) vleft = CHUNK;
    const int ntile = (vleft + 15) >> 4;
    const int vload = ntile << 4;

    __syncthreads();  // previous chunk's scans done before overwrite
    for (int i = tid; i < vload; i += 256) {
      float4 val;
      if (i < vleft) {
        const float* p = pv + ((size_t)b * VD + cb + i) * 3;
        const float vx = p[0], vy = p[1], vz = p[2];
        val = make_float4(vx, vy, vz, vx * vx + vy * vy + vz * vz);
      } else {
        val = make_float4(0.f, 0.f, 0.f, 3.0e38f);  // sentinel: never selected
      }
      vbuf[i] = val;
    }
    __syncthreads();

    // ---- two 16-vert tiles per iteration: 2 WMMAs, 1 fence ----
    const int npair = ntile >> 1;
    for (int tp = 0; tp < npair; ++tp) {
      const int t0 = tp << 1;
#pragma unroll
      for (int tt = 0; tt < 2; ++tt) {
        const float2* vp = (const float2*)(&vbuf[((t0 + tt) << 4) + m]);
        const float2 bv = vp[half];  // ds_load_b64: half0 -> (x,y), half1 -> (z,|v|^2)
        v2f Bv; Bv.x = bv.x; Bv.y = bv.y;
        v8f acc = {0.f, 0.f, 0.f, 0.f, 0.f, 0.f, 0.f, 0.f};
        // D[m][n] = |v_n|^2 - 2 p_m . v_n
        acc = __builtin_amdgcn_wmma_f32_16x16x4_f32(
            false, A, false, Bv, (short)0, acc, false, false);
#pragma unroll
        for (int r = 0; r < 8; ++r) {
          const int row = r + (half << 3);        // D layout: lanes16-31 hold M=r+8
          scr[w][tt][row * 17 + m] = acc[r];      // column index = lane%16
        }
      }
      wave_lds_fence();

      // batch all 16 candidate loads first (pipelined DS, partial waits),
      // then run the gated insertion on registers only
      float dv[16];
#pragma unroll
      for (int tt = 0; tt < 2; ++tt) {
#pragma unroll
        for (int j = 0; j < 8; ++j) {
          dv[tt * 8 + j] = scr[w][tt][m * 17 + (half << 3) + j];
        }
      }
#pragma unroll
      for (int tt = 0; tt < 2; ++tt) {
#pragma unroll
        for (int j = 0; j < 8; ++j) {
          const float d = dv[tt * 8 + j];
          // wave-uniform skip: insertion only if some lane improves its list
          if (__builtin_amdgcn_ballot_w32(d < bestd[KD - 1])) {
            insert6(bestd, besti, d,
                    cb + ((t0 + tt) << 4) + (half << 3) + j);
          }
        }
      }
    }
    // tail tile (odd ntile)
    if (ntile & 1) {
      const int t = ntile - 1;
      const float2* vp = (const float2*)(&vbuf[(t << 4) + m]);
      const float2 bv = vp[half];
      v2f Bv; Bv.x = bv.x; Bv.y = bv.y;
      v8f acc = {0.f, 0.f, 0.f, 0.f, 0.f, 0.f, 0.f, 0.f};
      acc = __builtin_amdgcn_wmma_f32_16x16x4_f32(
          false, A, false, Bv, (short)0, acc, false, false);
#pragma unroll
      for (int r = 0; r < 8; ++r) {
        const int row = r + (half << 3);
        scr[w][0][row * 17 + m] = acc[r];
      }
      wave_lds_fence();
      float dv[8];
#pragma unroll
      for (int j = 0; j < 8; ++j) {
        dv[j] = scr[w][0][m * 17 + (half << 3) + j];
      }
#pragma unroll
      for (int j = 0; j < 8; ++j) {
        const float d = dv[j];
        if (__builtin_amdgcn_ballot_w32(d < bestd[KD - 1])) {
          insert6(bestd, besti, d, cb + (t << 4) + (half << 3) + j);
        }
      }
    }
  }

  // ---- merge half-lists: lane L+16 -> lane L ----
  if (half == 1) {
#pragma unroll
    for (int s = 0; s < KD; ++s) {
      scr[w][0][m * 17 + s]     = bestd[s];
      scr[w][0][m * 17 + 8 + s] = __int_as_float(besti[s]);
    }
  }
  wave_lds_fence();
  if (half == 0) {
#pragma unroll
    for (int q = 0; q < KD; ++q) {
      const float d = scr[w][0][m * 17 + q];
      const int  gi = __float_as_int(scr[w][0][m * 17 + 8 + q]);
      insert6(bestd, besti, d, gi);
    }

    // ---- epilogue: conf, weights, blended rigid transform ----
    float dist[KD]; int idx[KD];
#pragma unroll
    for (int s = 0; s < KD; ++s) {
      dist[s] = fmaxf(bestd[s] + p2, 0.0f);       // true d^2, clamped like reference
      int ii = besti[s]; if (ii > VD - 1) ii = VD - 1;
      idx[s] = ii;
    }

    float sad[KD];
#pragma unroll
    for (int s = 0; s < KD; ++s) sad[s] = 0.0f;
    const float* w0p = lbs + (size_t)idx[0] * JD;
    for (int j = 0; j < JD; ++j) {
      const float w0 = w0p[j];
#pragma unroll
      for (int s = 1; s < KD; ++s)
        sad[s] += fabsf(lbs[(size_t)idx[s] * JD + j] - w0);
    }

    float wt[KD]; float wsum = 0.0f;
#pragma unroll
    for (int s = 0; s < KD; ++s) {
      const float conf = (s == 0) ? 1.0f
                       : ((expf(-sad[s] * 50.0f) > 0.9f) ? 1.0f : 0.0f); // /0.02
      wt[s] = expf(-dist[s]) * conf;
      wsum += wt[s];
    }
    const float invw = 1.0f / wsum;

    float T[12];
#pragma unroll
    for (int e = 0; e < 12; ++e) T[e] = 0.0f;
#pragma unroll
    for (int s = 0; s < KD; ++s) {
      const float ws = wt[s] * invw;
      const float* Tp = vt + (((size_t)b * VD + idx[s]) << 4);
#pragma unroll
      for (int e = 0; e < 12; ++e) T[e] += ws * Tp[e];
    }

    const int pt = ptbase + m;
    float* op = out + ((size_t)b * ND + pt) * 3;
    op[0] = T[0] * px + T[1] * py + T[2]  * pz + T[3];
    op[1] = T[4] * px + T[5] * py + T[6]  * pz + T[7];
    op[2] = T[8] * px + T[9] * py + T[10] * pz + T[11];
  }
}

extern "C" void kernel_launch(void* const* d_in, const int* in_sizes, int n_in,
                              void* d_out, int out_size, void* d_ws, size_t ws_size,
                              hipStream_t stream) {
  (void)in_sizes; (void)n_in; (void)out_size; (void)d_ws; (void)ws_size;
  const float* x   = (const float*)d_in[0];
  const float* cam = (const float*)d_in[1];
  const float* lbs = (const float*)d_in[2];
  const float* vt  = (const float*)d_in[3];
  const float* pv  = (const float*)d_in[4];
  float* out = (float*)d_out;

  dim3 grid(ND / PTS_PER_BLOCK, BD, 1);  // 128 x 2 blocks
  backdeform_knn_wmma_kernel<<<grid, 256, 0, stream>>>(x, cam, lbs, vt, pv, out);
}